// NoteCroppingsToPianorolls_13511967113921
// MI455X (gfx1250) — compile-verified
//
#include <hip/hip_runtime.h>

typedef __attribute__((ext_vector_type(2))) float v2f;
typedef __attribute__((ext_vector_type(8))) float v8f;

#define BATCH    8
#define NNOTES   512
#define TFRAMES  4096
#define NPITCH   88
#define NCLASS   11
#define HOP      512
#define NMAX     64               // max notes per (batch, pitch) bucket (binomial tail ~0)
#define KSTEPS   (NMAX / 4)       // WMMA k-chunks of 4 notes

// One workgroup per (batch, pitch) bucket. 8 wave32s; each wave computes 32
// time-tiles of 16 frames via V_WMMA_F32_16X16X4_F32 accumulation chains.
__global__ __launch_bounds__(256) void pianoroll_wmma_kernel(
    const int*   __restrict__ crops,   // (B, N, 3) = [pitch, start_idx, end_idx]
    const float* __restrict__ probs,   // (B, N, C)
    float*       __restrict__ out)     // (B, T, 88, C)
{
    __shared__ int  s_cnt;
    __shared__ int2 s_se[NMAX];            // (start_frame, end_frame) per slot
    __shared__ int  s_ni[NMAX];            // original note index per slot
    __shared__ v2f  s_bmat[KSTEPS][32];    // B-operand, per-lane WMMA layout

    const int blk = blockIdx.x;
    const int p   = blk % NPITCH;
    const int b   = blk / NPITCH;
    const int tid = threadIdx.x;

    if (tid == 0) s_cnt = 0;
    for (int i = tid; i < NMAX; i += 256) s_se[i] = make_int2(0, 0);
    __syncthreads();

    // ---- Bin valid notes of batch b with pitch p into LDS ----
    for (int n = tid; n < NNOTES; n += 256) {
        const int* nc = crops + ((size_t)b * NNOTES + n) * 3;
        const int pit   = nc[0] - 21;    // MIN_MIDI_PITCH
        const int e_idx = nc[2];
        if (pit == p && e_idx >= 0) {
            const int slot = atomicAdd(&s_cnt, 1);
            if (slot < NMAX) {
                s_ni[slot] = n;
                s_se[slot] = make_int2(nc[1] / HOP, e_idx / HOP);
            }
        }
    }
    __syncthreads();

    int cnt = s_cnt;
    if (cnt > NMAX) cnt = NMAX;
    const int ksteps = (cnt + 3) >> 2;     // uniform across the block

    // ---- Precompute B operand (4 notes x 16 classes) in WMMA lane layout ----
    // lane: N = lane&15 ; K pair = {kbase, kbase+1}, kbase = (lane>>4)*2
    for (int e = tid; e < KSTEPS * 32; e += 256) {
        const int k    = e >> 5;
        const int lane = e & 31;
        const int c    = lane & 15;
        const int kb   = (lane >> 4) << 1;
        const int n0   = 4 * k + kb;
        const int n1   = n0 + 1;
        v2f bv;
        bv.x = (n0 < cnt && c < NCLASS)
                 ? probs[((size_t)b * NNOTES + s_ni[n0]) * NCLASS + c] : 0.0f;
        bv.y = (n1 < cnt && c < NCLASS)
                 ? probs[((size_t)b * NNOTES + s_ni[n1]) * NCLASS + c] : 0.0f;
        s_bmat[k][lane] = bv;
    }
    __syncthreads();

    const int lane = tid & 31;
    const int wid  = tid >> 5;             // 8 waves
    const int m    = lane & 15;            // A-matrix row (frame within tile)
    const int kb   = (lane >> 4) << 1;     // K pair base (same for A and B)
    const int c    = lane & 15;            // output class column

    // 256 time-tiles of 16 frames; 32 tiles per wave
    for (int j = 0; j < 32; ++j) {
        const int tile  = wid * 32 + j;
        const int t0    = tile << 4;
        const int frame = t0 + m;

        v8f acc = {};
        for (int k = 0; k < ksteps; ++k) {            // uniform trip count
            const int2 se0 = s_se[4 * k + kb];
            const int2 se1 = s_se[4 * k + kb + 1];
            v2f a;
            a.x = (frame >= se0.x && frame < se0.y) ? 1.0f : 0.0f;
            a.y = (frame >= se1.x && frame < se1.y) ? 1.0f : 0.0f;
            // D = A(16x4 time-mask) * B(4x16 probs) + C
            acc = __builtin_amdgcn_wmma_f32_16x16x4_f32(
                      /*neg_a=*/false, a, /*neg_b=*/false, s_bmat[k][lane],
                      /*c_mod=*/(short)0, acc, /*reuse_a=*/false, /*reuse_b=*/false);
        }

        // C/D layout: lane col = lane&15; rows i + 8*(lane>=16) for VGPR i.
        if (c < NCLASS) {
            const size_t base  = (((size_t)b * TFRAMES + t0) * NPITCH + p) * NCLASS + c;
            const int    rbase = (lane >> 4) << 3;
            #pragma unroll
            for (int i = 0; i < 8; ++i) {
                out[base + (size_t)(rbase + i) * (NPITCH * NCLASS)] = acc[i];
            }
        }
    }
}

extern "C" void kernel_launch(void* const* d_in, const int* in_sizes, int n_in,
                              void* d_out, int out_size, void* d_ws, size_t ws_size,
                              hipStream_t stream) {
    (void)in_sizes; (void)n_in; (void)d_ws; (void)ws_size; (void)out_size;
    const int*   crops = (const int*)  d_in[0];   // note_croppings (8,512,3) int32
    const float* probs = (const float*)d_in[1];   // timbre_probs   (8,512,11) f32
    float*       out   = (float*)      d_out;     // (8,4096,88,11) f32, fully overwritten

    dim3 grid(BATCH * NPITCH);   // 704 buckets
    dim3 block(256);             // 8 wave32s
    pianoroll_wmma_kernel<<<grid, block, 0, stream>>>(crops, probs, out);
}